// SelfAttention_14379550507240
// MI455X (gfx1250) — compile-verified
//
#include <hip/hip_runtime.h>

typedef __attribute__((ext_vector_type(2))) float v2f;
typedef __attribute__((ext_vector_type(8))) float v8f;
typedef __attribute__((ext_vector_type(4))) int   v4i;

#define NB    2048
#define KV    1024
#define HD    64
#define NH    8
#define PD    6
#define EMB   512
#define CHUNK 64
#define NCHUNK (KV / CHUNK)
#define KSTR  68                       // k_tile row stride: 272B, 16B-aligned
#define SCALE 0.04419417382415922f     // 1/sqrt(512)

// ---- CDNA5 async Memory->LDS path (probe with fallback) -------------------
#if defined(__has_builtin)
#if __has_builtin(__builtin_amdgcn_global_load_async_to_lds_b128)
#define HAVE_ASYNC_LDS 1
#endif
#endif

#if __has_builtin(__builtin_amdgcn_s_wait_asynccnt)
#define WAIT_ASYNC(n) __builtin_amdgcn_s_wait_asynccnt(n)
#else
#define WAIT_ASYNC(n) asm volatile("s_wait_asynccnt %0" ::"n"(n))
#endif

#define GAS __attribute__((address_space(1)))
#define LAS __attribute__((address_space(3)))
typedef GAS v4i* gv4i_p;   // global-AS pointer to 16B beat
typedef LAS v4i* lv4i_p;   // LDS-AS pointer to 16B beat

// ---------------------------------------------------------------------------
// Kernel 1: fused low-rank-projected attention with online softmax.
// One workgroup (8 waves) per batch row; streams keys/values once (BW-bound).
// Keys are staged with double-buffered GLOBAL_LOAD_ASYNC_TO_LDS_B128 and
// retired with s_wait_asynccnt; attn @ values runs on v_wmma_f32_16x16x4_f32.
// ---------------------------------------------------------------------------
__global__ __launch_bounds__(256) void attn_kernel(
    const float* __restrict__ values, const float* __restrict__ keys,
    const float* __restrict__ queries, const float* __restrict__ W_k,
    const float* __restrict__ W_q, float* __restrict__ attn_out)
{
    __shared__ float k_tile[2][CHUNK][KSTR];  // double-buffered keys chunk
    __shared__ float k_projS[CHUNK][9];       // projected keys [64][6] (stride 9)
    __shared__ float p_tile[16][CHUNK];       // probabilities; rows 8..15 stay zero
    __shared__ float q_projS[NH][8];          // projected queries [8][6]
    __shared__ float alphaS[NH];              // online-softmax rescale per head
    __shared__ float lsumS[NH];               // final softmax denominators
    __shared__ float combineS[4][16][16];     // K-half accumulator combine buffer

    const int tid   = threadIdx.x;
    const int lane  = tid & 31;
    const int w     = tid >> 5;     // wave id 0..7
    const int n     = blockIdx.x;
    const int ntile = w & 3;        // which 16-col tile of HD=64
    const int khalf = w >> 2;       // which 32-row half of the chunk K dim

    const float* kb_g = keys   + (size_t)n * KV * HD;
    const float* vb_g = values + (size_t)n * KV * HD;

#if HAVE_ASYNC_LDS
    // stage one 64x64 f32 chunk: 4 x b128 async beats per thread, coalesced
    auto stage_chunk = [&](int c, int buf) {
        #pragma unroll
        for (int i = 0; i < 4; ++i) {
            int pos = tid + i * 256;               // float4 index within chunk
            int row = pos >> 4, c0 = (pos & 15) << 2;
            __builtin_amdgcn_global_load_async_to_lds_b128(
                (gv4i_p)(kb_g + (size_t)c * CHUNK * HD + (size_t)pos * 4),
                (lv4i_p)&k_tile[buf][row][c0], 0, 0);
        }
    };
#else
    auto stage_chunk = [&](int c, int buf) {
        const float4* kg = (const float4*)(kb_g + (size_t)c * CHUNK * HD);
        #pragma unroll
        for (int i = 0; i < 4; ++i) {
            int pos = tid + i * 256;
            float4 kv4 = kg[pos];
            int row = pos >> 4, c0 = (pos & 15) << 2;
            k_tile[buf][row][c0 + 0] = kv4.x;
            k_tile[buf][row][c0 + 1] = kv4.y;
            k_tile[buf][row][c0 + 2] = kv4.z;
            k_tile[buf][row][c0 + 3] = kv4.w;
        }
    };
#endif

    // zero the M-padding rows of the probability matrix (written once)
    for (int i = tid; i < 8 * CHUNK; i += 256)
        p_tile[8 + (i >> 6)][i & 63] = 0.0f;

    // q projection: [8 heads][64] @ [64][6]  (tiny, VALU)
    if (tid < NH * PD) {
        int h = tid / PD, p = tid % PD;
        const float* qr = queries + ((size_t)n * NH + h) * HD;
        const float* wr = W_q + p * HD;
        float s = 0.f;
        #pragma unroll
        for (int d = 0; d < HD; ++d) s += qr[d] * wr[d];
        q_projS[h][p] = s;
    }

    float M_run = -1e30f, L_run = 0.f;
    v8f acc = {};   // persistent WMMA accumulator (16x16 C tile)

#if HAVE_ASYNC_LDS
    stage_chunk(0, 0);   // prefetch chunk 0
#endif

    for (int c = 0; c < NCHUNK; ++c) {
        const int cur = c & 1;
        __syncthreads();   // p_tile/alphaS WAR + k_tile[next] WAR across chunks
#if HAVE_ASYNC_LDS
        if (c + 1 < NCHUNK) {
            stage_chunk(c + 1, cur ^ 1);   // prefetch next chunk
            WAIT_ASYNC(4);                 // retire current chunk's 4 beats
        } else {
            WAIT_ASYNC(0);
        }
#else
        stage_chunk(c, cur);
#endif
        __syncthreads();   // chunk c visible in LDS to all waves
        // ---- key projection: [64][64] @ [64][6] (tiny, VALU) ----
        if (tid < CHUNK * PD) {                  // 384 threads
            int p = tid >> 6, j = tid & 63;
            const float* wr = W_k + p * HD;
            float s = 0.f;
            #pragma unroll
            for (int d = 0; d < HD; ++d) s += k_tile[cur][j][d] * wr[d];
            k_projS[j][p] = s;
        }
        __syncthreads();
        // ---- energy + online softmax: wave w owns head w ----
        {
            float e0 = 0.f, e1 = 0.f;
            #pragma unroll
            for (int p = 0; p < PD; ++p) {
                float qv = q_projS[w][p];
                e0 += qv * k_projS[lane][p];
                e1 += qv * k_projS[lane + 32][p];
            }
            e0 *= SCALE; e1 *= SCALE;
            float m = fmaxf(e0, e1);
            #pragma unroll
            for (int off = 16; off > 0; off >>= 1)
                m = fmaxf(m, __shfl_xor(m, off, 32));
            float Mnew  = fmaxf(M_run, m);
            float alpha = __expf(M_run - Mnew);
            float p0 = __expf(e0 - Mnew);
            float p1 = __expf(e1 - Mnew);
            float ps = p0 + p1;
            #pragma unroll
            for (int off = 16; off > 0; off >>= 1)
                ps += __shfl_xor(ps, off, 32);
            L_run = L_run * alpha + ps;
            M_run = Mnew;
            p_tile[w][lane]      = p0;
            p_tile[w][lane + 32] = p1;
            if (lane == 0) alphaS[w] = alpha;
        }
        __syncthreads();
        // ---- rescale accumulator, then P[16x64] @ V[64x64] via fp32 WMMA ----
        {
            #pragma unroll
            for (int r = 0; r < 8; ++r) {
                // C layout: VGPR r = row r (lanes 0-15) / row r+8 (lanes 16-31).
                // Rows 8..15 are zero padding -> scale of 1 is fine there.
                float a = (lane < 16) ? alphaS[r] : 1.0f;
                acc[r] *= a;
            }
            const float* vchunk = vb_g + (size_t)c * CHUNK * HD;
            const int arow = lane & 15;
            const int bcol = ntile * 16 + (lane & 15);
            #pragma unroll
            for (int ks = 0; ks < 8; ++ks) {
                int kk = khalf * 32 + ks * 4 + ((lane >> 4) << 1);
                v2f afrag, bfrag;
                afrag.x = p_tile[arow][kk];
                afrag.y = p_tile[arow][kk + 1];
                bfrag.x = vchunk[(size_t)kk * HD + bcol];         // 64B coalesced
                bfrag.y = vchunk[(size_t)(kk + 1) * HD + bcol];
                acc = __builtin_amdgcn_wmma_f32_16x16x4_f32(
                          false, afrag, false, bfrag, (short)0, acc, false, false);
            }
        }
    }

    if (lane == 0) lsumS[w] = L_run;
    __syncthreads();
    // combine the two K-half accumulators, normalize, store [n][h*64+d]
    if (khalf == 1) {
        #pragma unroll
        for (int r = 0; r < 8; ++r)
            combineS[ntile][r + ((lane >> 4) << 3)][lane & 15] = acc[r];
    }
    __syncthreads();
    if (khalf == 0) {
        #pragma unroll
        for (int r = 0; r < 8; ++r)
            acc[r] += combineS[ntile][r + ((lane >> 4) << 3)][lane & 15];
        if (lane < 16) {   // only rows 0..7 (real heads) live in lanes 0-15
            float* orow = attn_out + (size_t)n * EMB;
            #pragma unroll
            for (int r = 0; r < 8; ++r)
                orow[r * HD + ntile * 16 + lane] = acc[r] / lsumS[r];
        }
    }
}

// ---------------------------------------------------------------------------
// Kernel 2: FC layer  C[2048,512] = A @ W^T + b  as LDS-tiled fp32 WMMA GEMM.
// 64x64 output tile per workgroup; each wave owns two 16x16 C tiles that share
// the B fragment. W_fc (1 MB) stays L2-resident across all workgroups.
// ---------------------------------------------------------------------------
__global__ __launch_bounds__(256) void fc_kernel(
    const float* __restrict__ A, const float* __restrict__ W,
    const float* __restrict__ bias, float* __restrict__ out)
{
    __shared__ float a_lds[64][33];
    __shared__ float w_lds[64][33];

    const int tid  = threadIdx.x;
    const int lane = tid & 31;
    const int w    = tid >> 5;
    const int rowBase = blockIdx.x * 64;
    const int colBase = blockIdx.y * 64;
    const int nt  = w & 3;          // col tile 0..3
    const int mt0 = w >> 2;         // row tiles mt0 and mt0+2
    const int mt1 = mt0 + 2;

    v8f acc0 = {}, acc1 = {};

    for (int kb = 0; kb < EMB; kb += 32) {
        __syncthreads();
        #pragma unroll
        for (int i = 0; i < 2; ++i) {                    // 64x32 floats each buffer
            int pos = tid + i * 256;
            int row = pos >> 3, c0 = (pos & 7) << 2;
            float4 av = *(const float4*)(A + (size_t)(rowBase + row) * EMB + kb + c0);
            a_lds[row][c0 + 0] = av.x; a_lds[row][c0 + 1] = av.y;
            a_lds[row][c0 + 2] = av.z; a_lds[row][c0 + 3] = av.w;
            float4 wv = *(const float4*)(W + (size_t)(colBase + row) * EMB + kb + c0);
            w_lds[row][c0 + 0] = wv.x; w_lds[row][c0 + 1] = wv.y;
            w_lds[row][c0 + 2] = wv.z; w_lds[row][c0 + 3] = wv.w;
        }
        __syncthreads();
        const int l15 = lane & 15;
        #pragma unroll
        for (int ks = 0; ks < 8; ++ks) {
            int kk = ks * 4 + ((lane >> 4) << 1);
            v2f bfrag, a0, a1;
            bfrag.x = w_lds[nt * 16 + l15][kk];          // B[k][n] = W[col][k]
            bfrag.y = w_lds[nt * 16 + l15][kk + 1];
            a0.x = a_lds[mt0 * 16 + l15][kk];
            a0.y = a_lds[mt0 * 16 + l15][kk + 1];
            a1.x = a_lds[mt1 * 16 + l15][kk];
            a1.y = a_lds[mt1 * 16 + l15][kk + 1];
            acc0 = __builtin_amdgcn_wmma_f32_16x16x4_f32(
                       false, a0, false, bfrag, (short)0, acc0, false, false);
            acc1 = __builtin_amdgcn_wmma_f32_16x16x4_f32(
                       false, a1, false, bfrag, (short)0, acc1, false, false);
        }
    }

    const int l15 = lane & 15;
    const int col = colBase + nt * 16 + l15;
    const float bv = bias[col];
    const int rsel = (lane >> 4) << 3;
    #pragma unroll
    for (int r = 0; r < 8; ++r) {
        out[(size_t)(rowBase + mt0 * 16 + r + rsel) * EMB + col] = acc0[r] + bv;
        out[(size_t)(rowBase + mt1 * 16 + r + rsel) * EMB + col] = acc1[r] + bv;
    }
}

extern "C" void kernel_launch(void* const* d_in, const int* in_sizes, int n_in,
                              void* d_out, int out_size, void* d_ws, size_t ws_size,
                              hipStream_t stream) {
    const float* values  = (const float*)d_in[0];
    const float* keys    = (const float*)d_in[1];
    const float* queries = (const float*)d_in[2];
    const float* W_k     = (const float*)d_in[3];
    const float* W_q     = (const float*)d_in[4];
    const float* W_fc    = (const float*)d_in[5];
    const float* b_fc    = (const float*)d_in[6];
    float* out     = (float*)d_out;
    float* attn_ws = (float*)d_ws;   // [2048][512] f32 = 4 MB intermediate

    attn_kernel<<<NB, 256, 0, stream>>>(values, keys, queries, W_k, W_q, attn_ws);
    fc_kernel<<<dim3(NB / 64, EMB / 64), 256, 0, stream>>>(attn_ws, W_fc, b_fc, out);
}